// EnergyModel_66331474919763
// MI455X (gfx1250) — compile-verified
//
#include <hip/hip_runtime.h>
#include <math.h>

// Compile-only target: MI455X / gfx1250 (wave32, WMMA).
// Energy model: bonds + angles + dihedrals + all-pairs LJ/Coulomb.
// d^2 tile computed almost entirely inside V_WMMA_F32_16X16X4_F32:
//   A row i = [-2x_i, -2y_i, -2z_i, sq_i], B col j = [x_j, y_j, z_j, 1]
//   => C[i][j] = sq_i - 2*p_i.p_j ; d2 = C + sq_j (one VALU add).
// Per-pair math: v_rsq_f32 + branch-free masked FMA accumulation.

#define KE2_C 0.000231f
#define EPS_C 1e-12f

typedef __attribute__((ext_vector_type(2))) float v2f;
typedef __attribute__((ext_vector_type(8))) float v8f;

__device__ __forceinline__ float norm3(float x, float y, float z) {
    // hardware v_sqrt_f32 -- avoids the IEEE refinement sequence
    return __builtin_amdgcn_sqrtf(x * x + y * y + z * z + EPS_C);
}

// blockDim.x == 256 (8 wave32) for all users.
__device__ __forceinline__ void block_reduce_add(float v, float* out) {
#pragma unroll
    for (int off = 16; off > 0; off >>= 1) v += __shfl_down(v, off, 32);
    __shared__ float wsum[8];
    int wave = threadIdx.x >> 5;
    int lane = threadIdx.x & 31;
    if (lane == 0) wsum[wave] = v;
    __syncthreads();
    if (threadIdx.x == 0) {
        float s = 0.0f;
#pragma unroll
        for (int w = 0; w < 8; ++w) s += wsum[w];
        atomicAdd(out, s);
    }
}

__global__ void init_kernel(float* out) {
    if (threadIdx.x == 0 && blockIdx.x == 0) out[0] = 0.0f;
}

__global__ void sq_kernel(const float* __restrict__ pos, float* __restrict__ sq, int n) {
    int i = blockIdx.x * blockDim.x + threadIdx.x;
    if (i < n) {
        float x = pos[i * 3 + 0];
        float y = pos[i * 3 + 1];
        float z = pos[i * 3 + 2];
        sq[i] = x * x + y * y + z * z;
    }
}

__global__ void __launch_bounds__(256)
bond_kernel(const float* __restrict__ pos, const int* __restrict__ bidx,
            const int* __restrict__ types, const float* __restrict__ bk,
            const float* __restrict__ bl, int nb, float* out) {
    int t = blockIdx.x * blockDim.x + threadIdx.x;
    float e = 0.0f;
    if (t < nb) {
        int a = bidx[t * 2 + 0];
        int b = bidx[t * 2 + 1];
        float dx = pos[a * 3 + 0] - pos[b * 3 + 0];
        float dy = pos[a * 3 + 1] - pos[b * 3 + 1];
        float dz = pos[a * 3 + 2] - pos[b * 3 + 2];
        float d = norm3(dx, dy, dz);
        int idx = types[a] * 7 + types[b];
        float dl = d - bl[idx];
        e = 0.5f * bk[idx] * dl * dl;
    }
    block_reduce_add(e, out);
}

__global__ void __launch_bounds__(256)
angle_kernel(const float* __restrict__ pos, const int* __restrict__ aidx,
             const int* __restrict__ types, const float* __restrict__ ak,
             const float* __restrict__ th, int na, float* out) {
    int t = blockIdx.x * blockDim.x + threadIdx.x;
    float e = 0.0f;
    if (t < na) {
        int i1 = aidx[t * 3 + 0];
        int i2 = aidx[t * 3 + 1];
        int i3 = aidx[t * 3 + 2];
        float v1x = pos[i1 * 3 + 0] - pos[i2 * 3 + 0];
        float v1y = pos[i1 * 3 + 1] - pos[i2 * 3 + 1];
        float v1z = pos[i1 * 3 + 2] - pos[i2 * 3 + 2];
        float v2x = pos[i3 * 3 + 0] - pos[i2 * 3 + 0];
        float v2y = pos[i3 * 3 + 1] - pos[i2 * 3 + 1];
        float v2z = pos[i3 * 3 + 2] - pos[i2 * 3 + 2];
        float c = (v1x * v2x + v1y * v2y + v1z * v2z) /
                  (norm3(v1x, v1y, v1z) * norm3(v2x, v2y, v2z));
        c = fminf(fmaxf(c, -1.0f + 1e-7f), 1.0f - 1e-7f);
        float ang = acosf(c);
        int idx = (types[i1] * 7 + types[i2]) * 7 + types[i3];
        float da = ang - th[idx];
        e = 0.5f * ak[idx] * da * da;
    }
    block_reduce_add(e, out);
}

__global__ void __launch_bounds__(256)
dihedral_kernel(const float* __restrict__ pos, const int* __restrict__ didx,
                const int* __restrict__ types, const float* __restrict__ dk,
                const float* __restrict__ mult, const float* __restrict__ ph,
                int nd, float* out) {
    int t = blockIdx.x * blockDim.x + threadIdx.x;
    float e = 0.0f;
    if (t < nd) {
        int i1 = didx[t * 4 + 0];
        int i2 = didx[t * 4 + 1];
        int i3 = didx[t * 4 + 2];
        int i4 = didx[t * 4 + 3];
        float w1x = pos[i2 * 3 + 0] - pos[i1 * 3 + 0];
        float w1y = pos[i2 * 3 + 1] - pos[i1 * 3 + 1];
        float w1z = pos[i2 * 3 + 2] - pos[i1 * 3 + 2];
        float w2x = pos[i3 * 3 + 0] - pos[i2 * 3 + 0];
        float w2y = pos[i3 * 3 + 1] - pos[i2 * 3 + 1];
        float w2z = pos[i3 * 3 + 2] - pos[i2 * 3 + 2];
        float w3x = pos[i4 * 3 + 0] - pos[i3 * 3 + 0];
        float w3y = pos[i4 * 3 + 1] - pos[i3 * 3 + 1];
        float w3z = pos[i4 * 3 + 2] - pos[i3 * 3 + 2];
        float n1x = w1y * w2z - w1z * w2y;
        float n1y = w1z * w2x - w1x * w2z;
        float n1z = w1x * w2y - w1y * w2x;
        float n2x = w2y * w3z - w2z * w3y;
        float n2y = w2z * w3x - w2x * w3z;
        float n2z = w2x * w3y - w2y * w3x;
        float nn1 = norm3(n1x, n1y, n1z);
        float nn2 = norm3(n2x, n2y, n2z);
        float cosd = (n1x * n2x + n1y * n2y + n1z * n2z) / (nn1 * nn2);
        float cxx = n1y * n2z - n1z * n2y;
        float cxy = n1z * n2x - n1x * n2z;
        float cxz = n1x * n2y - n1y * n2x;
        float sind = (cxx * w2x + cxy * w2y + cxz * w2z) /
                     (norm3(w2x, w2y, w2z) * nn1 * nn2);
        float phi = atan2f(sind, cosd);
        int idx = types[i2] * 7 + types[i3];
        e = 0.5f * dk[idx] * (1.0f + cosf(mult[idx] * phi - ph[idx]));
    }
    block_reduce_add(e, out);
}

// Pairwise LJ + Coulomb over the strict upper triangle; one 16x16 tile per wave.
__global__ void __launch_bounds__(256)
pair_kernel(const float* __restrict__ pos, const float* __restrict__ sq,
            const float* __restrict__ q, const int* __restrict__ ljt,
            const float* __restrict__ ljE, const float* __restrict__ ljR,
            int n, int T, int totalTiles, float* out) {
    __shared__ float sTab[32];  // interleaved (eps, sig) pairs -> one b64 load
    if (threadIdx.x < 16) {
        sTab[threadIdx.x * 2 + 0] = ljE[threadIdx.x];
        sTab[threadIdx.x * 2 + 1] = ljR[threadIdx.x];
    }
    __syncthreads();

    __builtin_prefetch(pos, 0, 1);  // global_prefetch_b8

    int wave = threadIdx.x >> 5;   // 8 waves (tiles) per block
    int lane = threadIdx.x & 31;
    int tile = blockIdx.x * 8 + wave;

    float acc = 0.0f;
    bool active = (tile < totalTiles);
    int I = 0, J = 0;
    if (active) {
        I = tile / T;
        J = tile - I * T;
        if (J < I) active = false;  // upper-triangle tiles only (wave-uniform)
    }

    if (active) {
        int half = lane >> 4;   // 0: K={0,1}; 1: K={2,3}
        int l = lane & 15;
        int rowA = I * 16 + l; if (rowA >= n) rowA = n - 1;
        int colB = J * 16 + l; if (colB >= n) colB = n - 1;

        float sqi_l = sq[rowA];

        // A = [-2x_i, -2y_i, -2z_i, sq_i] ; B = [x_j, y_j, z_j, 1]
        v2f a, b;
        if (half == 0) {
            a[0] = -2.0f * pos[rowA * 3 + 0]; a[1] = -2.0f * pos[rowA * 3 + 1];
            b[0] = pos[colB * 3 + 0];         b[1] = pos[colB * 3 + 1];
        } else {
            a[0] = -2.0f * pos[rowA * 3 + 2]; a[1] = sqi_l;
            b[0] = pos[colB * 3 + 2];         b[1] = 1.0f;
        }

        v8f c = {};
        // C[i][j] = sq_i - 2 * p_i . p_j
        c = __builtin_amdgcn_wmma_f32_16x16x4_f32(
                false, a, false, b, (short)0, c, false, false);

        // j-side values (per lane; N = lane & 15)
        int j = J * 16 + l;
        bool jok = (j < n);
        int jj = jok ? j : n - 1;
        float sqj = sq[jj];
        float qj = q[jj];
        int tj = ljt[jj];

        // i-side charge per row via cross-lane shuffle source
        float qi_l = q[rowA];
        // i-side lj type (2 bits) packed into wave-uniform ballot masks
        int ti_l = ljt[rowA];
        unsigned tb0 = __builtin_amdgcn_ballot_w32((ti_l & 1) != 0);
        unsigned tb1 = __builtin_amdgcn_ballot_w32((ti_l & 2) != 0);

        int mbase = half * 8;    // lanes 16-31 hold M = 8..15
        float accL = 0.0f;       // sum eps*(r6^2 - r6); *4 at end
        float accC = 0.0f;       // sum qi*qj/r;        *KE2 at end

#pragma unroll
        for (int r = 0; r < 8; ++r) {
            int src = mbase + r;
            int i = I * 16 + src;

            float qi = __shfl(qi_l, src, 32);
            int ti = (int)(((tb0 >> src) & 1u) | (((tb1 >> src) & 1u) << 1));

            bool m = (i < j) && jok;
            float fm = m ? 1.0f : 0.0f;

            float d2 = fmaxf(c[r] + sqj, 0.0f);
            float d2s = m ? d2 : 1.0f;   // keep masked lanes finite (no 0*Inf)
            float invr = __builtin_amdgcn_rsqf(d2s + EPS_C);

            int idx = (ti * 4 + tj) * 2;
            float epsij = sTab[idx + 0];
            float sig   = sTab[idx + 1];

            float s  = sig * invr;
            float s2 = s * s;
            float r6 = s2 * s2 * s2;
            accL = fmaf(fm * epsij, r6 * r6 - r6, accL);
            accC = fmaf(fm * qi, qj * invr, accC);
        }
        acc = 4.0f * accL + KE2_C * accC;
    }

    block_reduce_add(acc, out);
}

extern "C" void kernel_launch(void* const* d_in, const int* in_sizes, int n_in,
                              void* d_out, int out_size, void* d_ws, size_t ws_size,
                              hipStream_t stream) {
    const float* positions       = (const float*)d_in[0];
    const float* charges         = (const float*)d_in[1];
    const float* bond_coeffs     = (const float*)d_in[2];
    const float* bond_lengths    = (const float*)d_in[3];
    const float* angle_coeffs    = (const float*)d_in[4];
    const float* thetas          = (const float*)d_in[5];
    const float* dihedral_coeffs = (const float*)d_in[6];
    const float* phis            = (const float*)d_in[7];
    const float* multiplicity    = (const float*)d_in[8];
    const float* lj_E            = (const float*)d_in[9];
    const float* lj_r            = (const float*)d_in[10];
    const int*   atom_types      = (const int*)d_in[11];
    const int*   lj_types        = (const int*)d_in[12];
    const int*   bond_idx        = (const int*)d_in[13];
    const int*   angle_idx       = (const int*)d_in[14];
    const int*   dihedral_idx    = (const int*)d_in[15];

    float* out = (float*)d_out;
    float* sq  = (float*)d_ws;   // n floats of scratch

    int n  = in_sizes[1];        // 4096 atoms
    int nb = in_sizes[13] / 2;   // bonds
    int na = in_sizes[14] / 3;   // angles
    int nd = in_sizes[15] / 4;   // dihedrals

    init_kernel<<<1, 64, 0, stream>>>(out);
    sq_kernel<<<(n + 255) / 256, 256, 0, stream>>>(positions, sq, n);

    bond_kernel<<<(nb + 255) / 256, 256, 0, stream>>>(
        positions, bond_idx, atom_types, bond_coeffs, bond_lengths, nb, out);
    angle_kernel<<<(na + 255) / 256, 256, 0, stream>>>(
        positions, angle_idx, atom_types, angle_coeffs, thetas, na, out);
    dihedral_kernel<<<(nd + 255) / 256, 256, 0, stream>>>(
        positions, dihedral_idx, atom_types, dihedral_coeffs, multiplicity, phis,
        nd, out);

    int T = (n + 15) / 16;       // 256 tiles per side
    int totalTiles = T * T;
    int blocks = (totalTiles + 7) / 8;  // 8 waves (tiles) per 256-thread block
    pair_kernel<<<blocks, 256, 0, stream>>>(
        positions, sq, charges, lj_types, lj_E, lj_r, n, T, totalTiles, out);
}